// PairwiseScorer_60988535603681
// MI455X (gfx1250) — compile-verified
//
#include <hip/hip_runtime.h>
#include <hip/hip_bf16.h>

// PairwiseScorer on MI455X (gfx1250), fp32 WMMA path.
//
// Factorization:  x@W1 = g_q@W1[:512] (U, 16x96) + pair_feats@W1[512:] (V, Px96)
// K1: U = g_q@W1z + b1                      (tiny, VALU)
// K2: V = pair_feats@W1p                    (WMMA f32 16x16x4, feats built in LDS)
// K3: out = (gelu(gelu(U+V)@W2+b2)@W3+b3)   (fused, WMMA f32 16x16x4)
//
// Workspace: U (16*96 f32) then V (16384*96 f32)  ~6.3 MB.

#define B_DIM 16
#define P_DIM 16384
#define D_PHI 256
#define D_Z 512
#define D_R 64
#define H1 96
#define H2 96
#define PAIR_DIM (3 * D_PHI + D_R) /* 832 */

typedef __attribute__((ext_vector_type(2))) float v2f;
typedef __attribute__((ext_vector_type(8))) float v8f;

__device__ __forceinline__ v8f wmma4(v2f a, v2f b, v8f c) {
  // V_WMMA_F32_16X16X4_F32 : D(16x16 f32) = A(16x4 f32) * B(4x16 f32) + C
  return __builtin_amdgcn_wmma_f32_16x16x4_f32(false, a, false, b, (short)0, c,
                                               false, false);
}

__device__ __forceinline__ float gelu_exact(float x) {
  return 0.5f * x * (1.0f + erff(x * 0.7071067811865476f));
}

// ---------------- K1: U[b][c] = b1[c] + sum_k g_q[b][k] * W1[k][c] -----------
__global__ void __launch_bounds__(96)
u_kernel(const float* __restrict__ g_q, const float* __restrict__ W1,
         const float* __restrict__ b1, float* __restrict__ U) {
  const int c = threadIdx.x;
  if (c >= H1) return;
  for (int b = 0; b < B_DIM; ++b) {
    float acc = b1[c];
    const float* g = g_q + b * D_Z;
    for (int k = 0; k < D_Z; ++k) acc = fmaf(g[k], W1[k * H1 + c], acc);
    U[b * H1 + c] = acc;
  }
}

// ---------------- K2: V = pair_feats @ W1[512:, :] ---------------------------
// One block per 16-row p-tile, 96 threads = 3 waves, each wave 2 n-tiles.
#define SF 844 /* LDS row stride (floats): 844%64=12 -> conflict-free A frags */
__global__ void __launch_bounds__(96)
v_kernel(const float* __restrict__ Phi, const int* __restrict__ pair_index,
         const float* __restrict__ rel, const float* __restrict__ W1,
         float* __restrict__ V) {
  __shared__ float feats[16 * SF];
  __shared__ int idx[16][2];

  const int tid = threadIdx.x;
  const int p0 = blockIdx.x * 16;

  if (tid < 32) {
    const int r = tid >> 1, s = tid & 1;
    idx[r][s] = pair_index[(p0 + r) * 2 + s];
  }
  __syncthreads();

  // Build pair features (phi_i+phi_j | |phi_i-phi_j| | phi_i*phi_j | rel)
  for (int e = tid; e < 16 * D_PHI; e += 96) {
    const int r = e / D_PHI, c = e % D_PHI;
    const float pi = Phi[idx[r][0] * D_PHI + c];
    const float pj = Phi[idx[r][1] * D_PHI + c];
    float* row = &feats[r * SF];
    row[c] = pi + pj;
    row[D_PHI + c] = fabsf(pi - pj);
    row[2 * D_PHI + c] = pi * pj;
  }
  for (int e = tid; e < 16 * D_R; e += 96) {
    const int r = e / D_R, c = e % D_R;
    feats[r * SF + 3 * D_PHI + c] = rel[(size_t)(p0 + r) * D_R + c];
  }
  __syncthreads();

  const int wave = tid >> 5;
  const int lane = tid & 31;
  const int half = lane >> 4;
  const int l15 = lane & 15;

  for (int t = 0; t < 2; ++t) {
    const int n0 = (wave * 2 + t) * 16;
    v8f acc = {};
    for (int k0 = 0; k0 < PAIR_DIM; k0 += 4) {
      const int ka = k0 + half * 2;
      // A frag: lanes 0-15 rows, K={ka,ka+1}; lanes 16-31 K={ka,ka+1} (+2)
      const float* ap = &feats[l15 * SF + ka];
      v2f a;
      a.x = ap[0];
      a.y = ap[1];
      // B frag: lane = column, rows ka, ka+1 of W1p (row-major, stride H1)
      const float* bp = W1 + (size_t)(D_Z + ka) * H1 + n0 + l15;
      v2f bf;
      bf.x = bp[0];
      bf.y = bp[H1];
      acc = wmma4(a, bf, acc);
    }
    // C layout: VGPR v -> row m = v + 8*half, col n0 + l15
    for (int v = 0; v < 8; ++v) {
      const int m = v + 8 * half;
      V[(size_t)(p0 + m) * H1 + n0 + l15] = acc[v];
    }
  }
}

// ---------------- K3: fused gelu(U+V) @ W2 -> gelu -> @ W3 -------------------
// One block per 16-row p-tile, 128 threads = 4 waves, each wave 4 batch rows.
#define SW 104 /* W2 LDS stride: rows k and k+2 land on disjoint bank ranges  */
#define SV 100 /* V  LDS stride: 16 rows -> 16 distinct bank bases            */
__global__ void __launch_bounds__(128)
mlp_kernel(const float* __restrict__ U, const float* __restrict__ V,
           const float* __restrict__ W2, const float* __restrict__ b2,
           const float* __restrict__ W3, const float* __restrict__ b3,
           float* __restrict__ out) {
  __shared__ float W2s[H1 * SW];
  __shared__ float Us[B_DIM * H1];
  __shared__ float Vs[16 * SV];
  __shared__ float b2s[H2];
  __shared__ float w3s[H2];

  const int tid = threadIdx.x;
  const int p0 = blockIdx.x * 16;

  for (int e = tid; e < H1 * H2; e += 128)
    W2s[(e / H2) * SW + (e % H2)] = W2[e];
  for (int e = tid; e < B_DIM * H1; e += 128) Us[e] = U[e];
  for (int e = tid; e < 16 * H1; e += 128)
    Vs[(e / H1) * SV + (e % H1)] = V[(size_t)p0 * H1 + e];
  if (tid < H2) {
    b2s[tid] = b2[tid];
    w3s[tid] = W3[tid];
  }
  __syncthreads();

  const int wave = tid >> 5;
  const int lane = tid & 31;
  const int half = lane >> 4;
  const int l15 = lane & 15;
  const float bias3 = b3[0];
  const float* vrow = &Vs[l15 * SV];

  for (int i = 0; i < 4; ++i) {
    const int b = wave * 4 + i;
    const float* urow = &Us[b * H1];

    v8f acc[6];
#pragma unroll
    for (int nt = 0; nt < 6; ++nt) acc[nt] = (v8f){};

    for (int k0 = 0; k0 < H1; k0 += 4) {
      const int ka = k0 + half * 2;
      // A frag built on the fly: gelu(U[b][k] + V[row][k]) — layer-1 epilogue
      v2f a;
      a.x = gelu_exact(urow[ka] + vrow[ka]);
      a.y = gelu_exact(urow[ka + 1] + vrow[ka + 1]);
#pragma unroll
      for (int nt = 0; nt < 6; ++nt) {
        v2f bf;
        bf.x = W2s[ka * SW + nt * 16 + l15];
        bf.y = W2s[(ka + 1) * SW + nt * 16 + l15];
        acc[nt] = wmma4(a, bf, acc[nt]);
      }
    }

    // Layer-2 epilogue + layer-3 dot: each lane owns one n column per tile.
    float partial[8];
#pragma unroll
    for (int v = 0; v < 8; ++v) partial[v] = 0.0f;
#pragma unroll
    for (int nt = 0; nt < 6; ++nt) {
      const float bb = b2s[nt * 16 + l15];
      const float w3v = w3s[nt * 16 + l15];
#pragma unroll
      for (int v = 0; v < 8; ++v) {
        const float h2 = gelu_exact(acc[nt][v] + bb);
        partial[v] = fmaf(h2, w3v, partial[v]);
      }
    }
    // Reduce across the 16 lanes of each half (sum over n). xor<=8 stays
    // within each 16-lane half; halves hold disjoint row sets m=v / m=v+8.
#pragma unroll
    for (int off = 8; off >= 1; off >>= 1)
#pragma unroll
      for (int v = 0; v < 8; ++v)
        partial[v] += __shfl_xor(partial[v], off, 32);

    if (l15 == 0) {
#pragma unroll
      for (int v = 0; v < 8; ++v) {
        const int m = v + 8 * half;
        out[(size_t)b * P_DIM + p0 + m] = partial[v] + bias3;
      }
    }
  }
}

extern "C" void kernel_launch(void* const* d_in, const int* in_sizes, int n_in,
                              void* d_out, int out_size, void* d_ws,
                              size_t ws_size, hipStream_t stream) {
  const float* g_q = (const float*)d_in[0];
  const float* Phi = (const float*)d_in[1];
  const int* pair_index = (const int*)d_in[2];
  const float* rel = (const float*)d_in[3];
  const float* W1 = (const float*)d_in[4];
  const float* b1 = (const float*)d_in[5];
  const float* W2 = (const float*)d_in[6];
  const float* b2 = (const float*)d_in[7];
  const float* W3 = (const float*)d_in[8];
  const float* b3 = (const float*)d_in[9];
  float* out = (float*)d_out;

  float* Uws = (float*)d_ws;                 // 16*96 f32
  float* Vws = Uws + B_DIM * H1;             // 16384*96 f32

  u_kernel<<<1, 96, 0, stream>>>(g_q, W1, b1, Uws);
  v_kernel<<<P_DIM / 16, 96, 0, stream>>>(Phi, pair_index, rel, W1, Vws);
  mlp_kernel<<<P_DIM / 16, 128, 0, stream>>>(Uws, Vws, W2, b2, W3, b3, out);
}